// RoiPooling_21517786153630
// MI455X (gfx1250) — compile-verified
//
#include <hip/hip_runtime.h>
#include <stdint.h>
#include <stddef.h>

// ROI max pooling for MI455X (gfx1250, wave32).
// feature_map: [B,H,W,C] f32 NHWC, rois: [B,R,4] f32, out: [B,R,7,7,C] f32.
// Output quirk (from TF reference): out[b,r,i,j,c] = max over h-bin j, w-bin i.

typedef __attribute__((ext_vector_type(4))) float v4f;

namespace {
constexpr int kB = 4;
constexpr int kH = 64;
constexpr int kW = 64;
constexpr int kC = 256;
constexpr int kR = 128;
constexpr int kPH = 7;
constexpr int kPW = 7;
constexpr int kMaxPix = 25;           // bins are at most 5x5 pixels
constexpr int kThreads = 64;          // 2 waves; lane owns 4 channels (64*4 == 256)
}

__global__ __launch_bounds__(kThreads)
void roi_pool_gfx1250(const float* __restrict__ fm,
                      const float* __restrict__ rois,
                      float* __restrict__ out) {
  // One LDS slot (16B) per (pixel, lane): 25 * 64 * 16B = 25.6 KB.
  __shared__ v4f smem[kMaxPix * kThreads];

  const int tid = threadIdx.x;        // 0..63
  const int q   = blockIdx.x;         // 0..48 flattened output bin
  const int r   = blockIdx.y;
  const int b   = blockIdx.z;
  const int i   = q / kPW;            // output row  -> w-bin index (transpose quirk)
  const int j   = q - i * kPW;        // output col  -> h-bin index

  // ROI bounds (uniform per block; scalar loads).
  const float* roi = rois + (((size_t)b * kR) + r) * 4;
  const float x1 = roi[0], y1 = roi[1], x2 = roi[2], y2 = roi[3];
  int ws = max((int)rintf(x1), 0);            // jnp.round == round-half-even
  int hs = max((int)rintf(y1), 0);
  int we = min((int)rintf(x2), kW);
  int he = min((int)rintf(y2), kH);

  // TF bin bounds: float step, truncated-to-int offsets, last bin snaps to end.
  const float hstep = (float)(he - hs) / (float)kPH;
  const float wstep = (float)(we - ws) / (float)kPW;
  const int hlo = hs + (int)((float)j * hstep);
  const int hhi = (j + 1 < kPH) ? (hs + (int)((float)(j + 1) * hstep)) : he;
  const int wlo = ws + (int)((float)i * wstep);
  const int whi = (i + 1 < kPW) ? (ws + (int)((float)(i + 1) * wstep)) : we;

  int nh = hhi - hlo; nh = nh < 0 ? 0 : (nh > 5 ? 5 : nh);
  int nw = whi - wlo; nw = nw < 0 ? 0 : (nw > 5 ? 5 : nw);
  const int npix = nh * nw;

  // Per-lane LDS byte address base. Generic->LDS offset is the low 32 bits on AMDGPU.
  const uint32_t lds_base =
      (uint32_t)(uintptr_t)(&smem[0]) + (uint32_t)tid * 16u;

#if defined(__gfx1250__)
  // --- Stage all bin pixels into LDS with async copies (no dest VGPRs,
  //     up to 25 loads in flight per wave, tracked by ASYNCcnt only). ---
  {
    int p = 0;
    for (int hh = 0; hh < nh; ++hh) {
      const float* rowp = fm +
          ((((size_t)b * kH) + (size_t)(hlo + hh)) * kW + (size_t)wlo) * kC +
          4 * tid;
      for (int ww = 0; ww < nw; ++ww) {
        const uint64_t gaddr = (uint64_t)(uintptr_t)(rowp + (size_t)ww * kC);
        const uint32_t laddr = lds_base + (uint32_t)p * (kThreads * 16u);
        asm volatile("global_load_async_to_lds_b128 %0, %1, off"
                     :: "v"(laddr), "v"(gaddr)
                     : "memory");
        ++p;
      }
    }
  }
  asm volatile("s_wait_asynccnt 0x0" ::: "memory");

  // Reduce from LDS: each lane reads back exactly the slots its wave wrote,
  // so ASYNCcnt==0 is sufficient (no workgroup barrier needed).
  v4f acc = {-INFINITY, -INFINITY, -INFINITY, -INFINITY};
  for (int p = 0; p < npix; ++p) {
    v4f v = smem[p * kThreads + tid];
    acc.x = fmaxf(acc.x, v.x);
    acc.y = fmaxf(acc.y, v.y);
    acc.z = fmaxf(acc.z, v.z);
    acc.w = fmaxf(acc.w, v.w);
  }
#else
  // Fallback: direct vector loads.
  v4f acc = {-INFINITY, -INFINITY, -INFINITY, -INFINITY};
  for (int hh = 0; hh < nh; ++hh) {
    const float* rowp = fm +
        ((((size_t)b * kH) + (size_t)(hlo + hh)) * kW + (size_t)wlo) * kC +
        4 * tid;
    for (int ww = 0; ww < nw; ++ww) {
      v4f v = *(const v4f*)(rowp + (size_t)ww * kC);
      acc.x = fmaxf(acc.x, v.x);
      acc.y = fmaxf(acc.y, v.y);
      acc.z = fmaxf(acc.z, v.z);
      acc.w = fmaxf(acc.w, v.w);
    }
  }
#endif

  // out[b, r, i, j, c] ; flat bin index is exactly q. Non-temporal store:
  // output is streamed once, keep the feature map resident in L2.
  float* op = out +
      ((((size_t)b * kR) + r) * (kPH * kPW) + (size_t)q) * kC + 4 * tid;
  __builtin_nontemporal_store(acc, (v4f*)op);
}

extern "C" void kernel_launch(void* const* d_in, const int* in_sizes, int n_in,
                              void* d_out, int out_size, void* d_ws, size_t ws_size,
                              hipStream_t stream) {
  (void)in_sizes; (void)n_in; (void)d_ws; (void)ws_size; (void)out_size;
  const float* fm   = (const float*)d_in[0];
  const float* rois = (const float*)d_in[1];
  float* out        = (float*)d_out;

  dim3 grid(kPH * kPW, kR, kB);   // (49, 128, 4) blocks
  roi_pool_gfx1250<<<grid, kThreads, 0, stream>>>(fm, rois, out);
}